// Rwkv6SelfAttention_61564061221494
// MI455X (gfx1250) — compile-verified
//
#include <hip/hip_runtime.h>
#include <math.h>

#define H   4096
#define HS  64
#define NH  64
#define TM  64
#define TD  128
#define EPSV 1e-5f

typedef __attribute__((ext_vector_type(2))) float v2f;
typedef __attribute__((ext_vector_type(8))) float v8f;

__device__ __forceinline__ float wave_sum32(float v) {
    for (int m = 16; m > 0; m >>= 1) v += __shfl_xor(v, m, 32);
    return v;
}

// ---------------- K1: LayerNorm(x), token-shift delta, LoRA input --------------
__global__ void k_ln_mix(const float* __restrict__ x, const float* __restrict__ st1,
                         const float* __restrict__ w, const float* __restrict__ b,
                         const float* __restrict__ tmx,
                         float* __restrict__ xl_out, float* __restrict__ xl_ws,
                         float* __restrict__ sx_ws, float* __restrict__ xin_ws) {
    __shared__ float ssum[256], ssq[256];
    int tid = threadIdx.x;
    float s = 0.f, q = 0.f;
    for (int i = tid; i < H; i += 256) { float v = x[i]; s += v; q += v * v; }
    ssum[tid] = s; ssq[tid] = q; __syncthreads();
    for (int o = 128; o > 0; o >>= 1) {
        if (tid < o) { ssum[tid] += ssum[tid + o]; ssq[tid] += ssq[tid + o]; }
        __syncthreads();
    }
    float mean = ssum[0] * (1.f / H);
    float var  = ssq[0] * (1.f / H) - mean * mean;
    float inv  = rsqrtf(var + EPSV);
    for (int i = tid; i < H; i += 256) {
        float xv  = x[i];
        float xln = (xv - mean) * inv * w[i] + b[i];
        float sx  = st1[i] - xln;
        xl_out[i] = xln; xl_ws[i] = xln; sx_ws[i] = sx;
        xin_ws[i] = xln + sx * tmx[i];
    }
}

// ---------------- K2: LoRA down xin @ maa_w1 (partials over h-chunks) ----------
__global__ void k_maa_down(const float* __restrict__ xin, const float* __restrict__ w1,
                           float* __restrict__ part) {
    int c  = threadIdx.x;          // 0..319
    int h0 = blockIdx.x * 256;
    float s = 0.f;
    for (int h = h0; h < h0 + 256; ++h) s += xin[h] * w1[h * 320 + c];
    part[blockIdx.x * 320 + c] = s;
}

// ---------------- K3: finalize tanh, LoRA up, form the 5 mix vectors -----------
__global__ void k_maa_up(const float* __restrict__ part, const float* __restrict__ w2,
                         const float* __restrict__ xl, const float* __restrict__ sx,
                         const float* __restrict__ tmaa, float* __restrict__ mix) {
    __shared__ float tl[TM];
    int f   = blockIdx.x / 16;
    int hh0 = (blockIdx.x % 16) * 256;
    int tid = threadIdx.x;
    if (tid < TM) {
        float s = 0.f;
        for (int p = 0; p < 16; ++p) s += part[p * 320 + f * TM + tid];
        tl[tid] = tanhf(s);
    }
    __syncthreads();
    int hh = hh0 + tid;
    float acc = 0.f;
    const float* w2f = w2 + (size_t)f * TM * H;
    for (int t = 0; t < TM; ++t) acc += tl[t] * w2f[t * H + hh];
    mix[f * H + hh] = xl[hh] + sx[hh] * (acc + tmaa[f * H + hh]);
}

// ---------------- K4: the four big GEMVs (bandwidth-bound), wave per row -------
__global__ void k_rkvg(const float* __restrict__ Wr, const float* __restrict__ Wk,
                       const float* __restrict__ Wv, const float* __restrict__ Wg,
                       const float* __restrict__ mix,
                       float* __restrict__ rv, float* __restrict__ kv,
                       float* __restrict__ vv, float* __restrict__ gv) {
    __shared__ float vec[H];
    int mat = blockIdx.x >> 9;       // 0:r 1:k 2:v 3:g (512 blocks each)
    const float* W; const float* m; float* out;
    if      (mat == 0) { W = Wr; m = mix + 3 * H; out = rv; }
    else if (mat == 1) { W = Wk; m = mix + 1 * H; out = kv; }
    else if (mat == 2) { W = Wv; m = mix + 2 * H; out = vv; }
    else               { W = Wg; m = mix + 4 * H; out = gv; }
    for (int i = threadIdx.x; i < H; i += 256) vec[i] = m[i];
    __syncthreads();
    int wave = threadIdx.x >> 5, lane = threadIdx.x & 31;
    int row  = ((blockIdx.x & 511) << 3) + wave;
    const float* wr = W + (size_t)row * H;
    float acc = 0.f;
    for (int t = lane * 4; t < H; t += 128) {
        float4 w4 = *reinterpret_cast<const float4*>(wr + t);
        acc += w4.x * vec[t] + w4.y * vec[t + 1] + w4.z * vec[t + 2] + w4.w * vec[t + 3];
    }
    acc = wave_sum32(acc);
    if (lane == 0) {
        if (mat == 3) acc = acc / (1.f + __expf(-acc));   // fused SiLU for g
        out[row] = acc;
    }
}

// ---------------- K5/K6: data-dependent decay LoRA -----------------------------
__global__ void k_dec_down(const float* __restrict__ mw, const float* __restrict__ w1,
                           float* __restrict__ part) {
    int c  = threadIdx.x;          // 0..127
    int h0 = blockIdx.x * 256;
    float s = 0.f;
    for (int h = h0; h < h0 + 256; ++h) s += mw[h] * w1[h * TD + c];
    part[blockIdx.x * TD + c] = s;
}

__global__ void k_dec_up(const float* __restrict__ part, const float* __restrict__ w2,
                         const float* __restrict__ tdcy, float* __restrict__ wdec) {
    __shared__ float tl[TD];
    int tid = threadIdx.x;
    if (tid < TD) {
        float s = 0.f;
        for (int p = 0; p < 16; ++p) s += part[p * TD + tid];
        tl[tid] = tanhf(s);
    }
    __syncthreads();
    int i = blockIdx.x * 256 + tid;
    float acc = 0.f;
    for (int c = 0; c < TD; ++c) acc += tl[c] * w2[c * H + i];
    float td = acc + tdcy[i];
    td = fminf(fmaxf(td, -9.72f), 2.27f);
    wdec[i] = __expf(-__expf(td));
}

// ---------------- K7: state recurrence state2' = k v^T + state2 * decay --------
__global__ void k_state(const float* __restrict__ kvec, const float* __restrict__ vvec,
                        const float* __restrict__ S, const float* __restrict__ wdec,
                        float* __restrict__ s2out) {
    int idx = blockIdx.x * 256 + threadIdx.x;   // 0..262143
    int h = idx >> 12, i = (idx >> 6) & 63, j = idx & 63;
    s2out[idx] = kvec[h * 64 + i] * vvec[h * 64 + j] + S[idx] * wdec[h * 64 + i];
}

// ---------------- K8: per-head readout via V_WMMA_F32_16X16X4_F32 --------------
// out = r @ (tf*k⊗v + S), then InstanceNorm + affine + gate, one wave per head.
// A-fragment reads are UNCONDITIONAL LDS loads zeroed by a per-lane mask, so the
// compiler emits clean ds_load -> v_wmma sequences without exec save/restore.
__global__ void __launch_bounds__(32) k_head(
    const float* __restrict__ rvec, const float* __restrict__ kvec,
    const float* __restrict__ vvec, const float* __restrict__ tf,
    const float* __restrict__ S, const float* __restrict__ gvec,
    const float* __restrict__ lnw, const float* __restrict__ lnb,
    float* __restrict__ og) {
    __shared__ float rl[64], al[64], vl[64], outl[64];
    int h = blockIdx.x;
    int lane = threadIdx.x;
    for (int i = lane; i < 64; i += 32) {
        rl[i] = rvec[h * 64 + i];
        al[i] = tf[h * 64 + i] * kvec[h * 64 + i];
        vl[i] = vvec[h * 64 + i];
    }
    __syncthreads();
    const float* Sh = S + (size_t)h * 4096;
    int col  = lane & 15;                      // N column within tile / A row id
    int half = lane >> 4;                      // 0: K pair {0,1}, 1: K pair {2,3}
    const float rmask = (col == 0) ? 1.f : 0.f; // only M=0 row of A carries r
    for (int nt = 0; nt < 4; ++nt) {
        v8f acc = {0.f, 0.f, 0.f, 0.f, 0.f, 0.f, 0.f, 0.f};
        int j = nt * 16 + col;
        float vj = vl[j];
        for (int kc = 0; kc < 16; ++kc) {
            int kr = 4 * kc + half * 2;
            // unconditional LDS loads (in-bounds for every lane), masked by rmask
            float r0 = rl[kr], r1 = rl[kr + 1];
            float a0 = al[kr], a1 = al[kr + 1];
            v2f a, b;
            // A 16x4 layout: lanes 0-15 rows M, vgpr0=K0 (K2 for lanes>=16), vgpr1=K1/K3
            a.x = r0 * rmask;
            a.y = r1 * rmask;
            // B 4x16 tile of M = tf*k[i]*v[j] + S[i][j]
            b.x = a0 * vj + Sh[kr * 64 + j];
            b.y = a1 * vj + Sh[(kr + 1) * 64 + j];
#if __has_builtin(__builtin_amdgcn_wmma_f32_16x16x4_f32)
            acc = __builtin_amdgcn_wmma_f32_16x16x4_f32(
                false, a, false, b, (short)0, acc, false, false);
#else
            acc[0] += r0 * b.x + r1 * b.y;
#endif
        }
#if __has_builtin(__builtin_amdgcn_wmma_f32_16x16x4_f32)
        if (lane < 16) outl[nt * 16 + lane] = acc[0];   // D row 0 = vgpr0, lanes 0-15
#else
        float tot = acc[0] + __shfl_xor(acc[0], 16, 32);
        if (lane < 16) outl[nt * 16 + lane] = tot;
#endif
    }
    __syncthreads();
    float v0 = outl[lane], v1 = outl[lane + 32];
    float s = wave_sum32(v0 + v1);
    float q = wave_sum32(v0 * v0 + v1 * v1);
    float mean = s * (1.f / 64.f);
    float var  = q * (1.f / 64.f) - mean * mean;
    float inv  = rsqrtf(var + EPSV);
    int c0 = h * 64 + lane, c1 = c0 + 32;
    og[c0] = ((v0 - mean) * inv * lnw[c0] + lnb[c0]) * gvec[c0];
    og[c1] = ((v1 - mean) * inv * lnw[c1] + lnb[c1]) * gvec[c1];
}

// ---------------- K9: final GEMV  new_x = x + og @ Wo^T ------------------------
__global__ void k_out(const float* __restrict__ Wo, const float* __restrict__ og,
                      const float* __restrict__ x, float* __restrict__ outx) {
    __shared__ float vec[H];
    for (int i = threadIdx.x; i < H; i += 256) vec[i] = og[i];
    __syncthreads();
    int wave = threadIdx.x >> 5, lane = threadIdx.x & 31;
    int row  = (blockIdx.x << 3) + wave;
    const float* wr = Wo + (size_t)row * H;
    float acc = 0.f;
    for (int t = lane * 4; t < H; t += 128) {
        float4 w4 = *reinterpret_cast<const float4*>(wr + t);
        acc += w4.x * vec[t] + w4.y * vec[t + 1] + w4.z * vec[t + 2] + w4.w * vec[t + 3];
    }
    acc = wave_sum32(acc);
    if (lane == 0) outx[row] = x[row] + acc;
}

extern "C" void kernel_launch(void* const* d_in, const int* in_sizes, int n_in,
                              void* d_out, int out_size, void* d_ws, size_t ws_size,
                              hipStream_t stream) {
    const float* x     = (const float*)d_in[0];
    const float* st1   = (const float*)d_in[1];
    const float* st2   = (const float*)d_in[2];
    const float* ln1w  = (const float*)d_in[3];
    const float* ln1b  = (const float*)d_in[4];
    const float* tmx   = (const float*)d_in[5];
    const float* tmaa  = (const float*)d_in[6];
    const float* maw1  = (const float*)d_in[7];
    const float* maw2  = (const float*)d_in[8];
    const float* tdcy  = (const float*)d_in[9];
    const float* tfst  = (const float*)d_in[10];
    const float* dw1   = (const float*)d_in[11];
    const float* dw2   = (const float*)d_in[12];
    const float* Wr    = (const float*)d_in[13];
    const float* Wk    = (const float*)d_in[14];
    const float* Wv    = (const float*)d_in[15];
    const float* Wg    = (const float*)d_in[16];
    const float* Wo    = (const float*)d_in[17];
    const float* lnxw  = (const float*)d_in[18];
    const float* lnxb  = (const float*)d_in[19];

    float* out    = (float*)d_out;
    float* o_newx = out;            // [0:4096)
    float* o_xl   = out + H;        // [4096:8192)
    float* o_st2  = out + 2 * H;    // [8192:270336)

    float* ws = (float*)d_ws;
    float* w_xl   = ws;                 // 4096
    float* w_sx   = w_xl   + H;         // 4096
    float* w_xin  = w_sx   + H;         // 4096
    float* w_mpart= w_xin  + H;         // 16*320
    float* w_mix  = w_mpart+ 16 * 320;  // 5*4096
    float* w_r    = w_mix  + 5 * H;     // 4096
    float* w_k    = w_r    + H;         // 4096
    float* w_v    = w_k    + H;         // 4096
    float* w_g    = w_v    + H;         // 4096
    float* w_dpart= w_g    + H;         // 16*128
    float* w_wdec = w_dpart+ 16 * TD;   // 4096
    float* w_og   = w_wdec + H;         // 4096

    k_ln_mix<<<1, 256, 0, stream>>>(x, st1, ln1w, ln1b, tmx, o_xl, w_xl, w_sx, w_xin);
    k_maa_down<<<16, 320, 0, stream>>>(w_xin, maw1, w_mpart);
    k_maa_up<<<80, 256, 0, stream>>>(w_mpart, maw2, w_xl, w_sx, tmaa, w_mix);
    k_rkvg<<<2048, 256, 0, stream>>>(Wr, Wk, Wv, Wg, w_mix, w_r, w_k, w_v, w_g);
    k_dec_down<<<16, 128, 0, stream>>>(w_mix /* mix[0] = mw */, dw1, w_dpart);
    k_dec_up<<<16, 256, 0, stream>>>(w_dpart, dw2, tdcy, w_wdec);
    k_state<<<1024, 256, 0, stream>>>(w_k, w_v, st2, w_wdec, o_st2);
    k_head<<<NH, 32, 0, stream>>>(w_r, w_k, w_v, tfst, st2, w_g, lnxw, lnxb, w_og);
    k_out<<<512, 256, 0, stream>>>(Wo, w_og, x, o_newx);
}